// soft_edp_loss_38783554683053
// MI455X (gfx1250) — compile-verified
//
#include <hip/hip_runtime.h>

// soft EDP loss for MI455X (gfx1250): wave32, WMMA f32 16x16x4, async global->LDS.
//
// L = 48000 outputs per signal, window = 1025 taps.
// Block = 256 threads (8 waves), each wave computes 16 consecutive outputs for
// BOTH signals; block tile = 128 outputs -> 375 blocks exactly.
// Needed inputs per block: x[l0-1025 .. l0+126] = 1152 floats per signal,
// staged in LDS (async global->LDS on the interior blocks).

#define L_SIG        48000
#define WLEN         1025
#define ROWS_PB      128
#define THREADS      256
#define NCHUNK       257      // ceil(1025/4) K-chunks of 4
#define XTAIL        1156     // 1152 valid + 4 zero pad (k up to 1027)
#define WEXT         1028     // 1025 window taps + 3 zeros

typedef __attribute__((ext_vector_type(2))) float v2f;
typedef __attribute__((ext_vector_type(8))) float v8f;

__device__ __forceinline__ float fast_sigmoid(float t) {
  // 1/(1+e^-t) = 1/(1 + 2^(-t*log2(e)))
  float e = __builtin_amdgcn_exp2f(-t * 1.44269504088896f);
  return __builtin_amdgcn_rcpf(1.0f + e);
}

__global__ __launch_bounds__(THREADS) void edp_loss_kernel(
    const float* __restrict__ yp, const float* __restrict__ yt,
    const float* __restrict__ win, float* __restrict__ partials) {
  __shared__ float xs[2][XTAIL];
  __shared__ float wext[WEXT];
  __shared__ float wavesum[16];

  const int tid = threadIdx.x;
  const int l0 = blockIdx.x * ROWS_PB;
  const long gbase = (long)l0 - WLEN;   // global index of xs[s][0]

  // window + zero tail into LDS
  for (int i = tid; i < WEXT; i += THREADS) wext[i] = (i < WLEN) ? win[i] : 0.0f;
  if (tid < 4) { xs[0][1152 + tid] = 0.0f; xs[1][1152 + tid] = 0.0f; }

  const float* src0 = yp;
  const float* src1 = yt;

  if (gbase >= 0) {
    // Interior block: CDNA5 async global->LDS (ASYNCcnt-tracked DMA path).
    unsigned lds0 = (unsigned)(size_t)(void*)&xs[0][0];
    unsigned lds1 = (unsigned)(size_t)(void*)&xs[1][0];
#pragma unroll
    for (int i = 0; i < 5; ++i) {
      int idx = i * THREADS + tid;
      if (idx < 1152) {
        const float* g0 = src0 + gbase + idx;
        const float* g1 = src1 + gbase + idx;
        asm volatile("global_load_async_to_lds_b32 %0, %1, off"
                     :: "v"(lds0 + 4u * (unsigned)idx), "v"(g0) : "memory");
        asm volatile("global_load_async_to_lds_b32 %0, %1, off"
                     :: "v"(lds1 + 4u * (unsigned)idx), "v"(g1) : "memory");
      }
    }
    asm volatile("s_wait_asynccnt 0x0" ::: "memory");
  } else {
    // First 9 blocks touch the implicit left zero-padding: predicated loads.
#pragma unroll
    for (int i = 0; i < 5; ++i) {
      int idx = i * THREADS + tid;
      if (idx < 1152) {
        long g = gbase + idx;
        xs[0][idx] = (g >= 0) ? src0[g] : 0.0f;
        xs[1][idx] = (g >= 0) ? src1[g] : 0.0f;
      }
    }
  }
  __syncthreads();

  const int lane = tid & 31;
  const int wave = tid >> 5;
  const int r    = lane & 15;     // row of the wave's 16x? tile this lane serves
  const int h    = lane >> 4;     // half-wave: selects K pair (A) / row group (D)
  const int rowInBlk = wave * 16 + r;
  const int l = l0 + rowInBlk;
  const float kappa = 100.0f + 2.08125f * (float)l;   // (1e5-1e2)/48000

  v8f acc_p, acc_t;
#pragma unroll
  for (int s = 0; s < 2; ++s) {
    const float* xp = xs[s];

    // --- frame std (ddof=1): half-wave partial sums, combined via shfl_xor 16
    float sum = 0.0f, sq = 0.0f;
    const int kbeg = h ? 513 : 0;
    const int kend = h ? 1025 : 513;
#pragma unroll 4
    for (int k = kbeg; k < kend; ++k) {
      float v = xp[rowInBlk + k];
      sum += v;
      sq  = fmaf(v, v, sq);
    }
    sum += __shfl_xor(sum, 16, 32);
    sq  += __shfl_xor(sq, 16, 32);
    float mean = sum * (1.0f / 1025.0f);
    float var  = (sq - 1025.0f * mean * mean) * (1.0f / 1024.0f);
    float stdv = sqrtf(fmaxf(var, 0.0f));

    // --- windowed sigmoid sum via V_WMMA_F32_16X16X4_F32
    // A[m][k] = sigmoid(|x*kappa_m| - std_m); B[k][n] = window[k] (all columns)
    // lane L: A vgpr0/1 = K = 4c + 2h {+0,+1}; B vgpr0/1 = window at same K.
    v8f acc = {0.f, 0.f, 0.f, 0.f, 0.f, 0.f, 0.f, 0.f};
    for (int c = 0; c < NCHUNK; ++c) {
      const int kA = 4 * c + 2 * h;
      float x0 = xp[rowInBlk + kA];
      float x1 = xp[rowInBlk + kA + 1];
      v2f a, b;
      a.x = fast_sigmoid(fabsf(x0 * kappa) - stdv);
      a.y = fast_sigmoid(fabsf(x1 * kappa) - stdv);
      b.x = wext[kA];
      b.y = wext[kA + 1];
      acc = __builtin_amdgcn_wmma_f32_16x16x4_f32(
          false, a, false, b, (short)0, acc, false, false);
    }
    if (s == 0) acc_p = acc; else acc_t = acc;
  }

  // D columns are identical (B broadcast): lanes with r==0 (lanes 0 and 16)
  // hold rows h*8 .. h*8+7 in acc[0..7]. Fixed-order partial per block.
  const float inv_erfc = 3.1514873f;   // 1/erfc(1/sqrt(2))
  if (r == 0) {
    float p = 0.0f;
#pragma unroll
    for (int i = 0; i < 8; ++i) {
      float d = (acc_t[i] - acc_p[i]) * inv_erfc;
      p = fmaf(d, d, p);
    }
    wavesum[wave * 2 + h] = p;
  }
  __syncthreads();
  if (tid == 0) {
    float t = 0.0f;
#pragma unroll
    for (int i = 0; i < 16; ++i) t += wavesum[i];
    partials[blockIdx.x] = t;
  }
}

__global__ __launch_bounds__(256) void edp_reduce_kernel(
    const float* __restrict__ partials, float* __restrict__ out, int n) {
  __shared__ float sm[256];
  float s = 0.0f;
  for (int i = threadIdx.x; i < n; i += 256) s += partials[i];
  sm[threadIdx.x] = s;
  __syncthreads();
  for (int off = 128; off > 0; off >>= 1) {
    if (threadIdx.x < off) sm[threadIdx.x] += sm[threadIdx.x + off];
    __syncthreads();
  }
  if (threadIdx.x == 0) out[0] = sm[0] * (1.0f / (float)L_SIG);
}

extern "C" void kernel_launch(void* const* d_in, const int* in_sizes, int n_in,
                              void* d_out, int out_size, void* d_ws, size_t ws_size,
                              hipStream_t stream) {
  const float* yp  = (const float*)d_in[0];   // (1, 48000, 1) f32
  const float* yt  = (const float*)d_in[1];   // (1, 48000, 1) f32
  const float* win = (const float*)d_in[2];   // (1025,) f32
  float* partials  = (float*)d_ws;            // 375 floats
  float* out       = (float*)d_out;           // scalar f32

  const int nblocks = L_SIG / ROWS_PB;        // 375 exactly
  edp_loss_kernel<<<nblocks, THREADS, 0, stream>>>(yp, yt, win, partials);
  edp_reduce_kernel<<<1, 256, 0, stream>>>(partials, out, nblocks);
}